// DescriptorMatcher_55181739819638
// MI455X (gfx1250) — compile-verified
//
#include <hip/hip_runtime.h>

typedef float v2f __attribute__((ext_vector_type(2)));
typedef float v4f __attribute__((ext_vector_type(4)));
typedef float v8f __attribute__((ext_vector_type(8)));

#define DIM 256

// ---------------------------------------------------------------------------
// Pass 1: column norms ||desc2[n]||^2 -> b2[n]  (64 KB, L2/L0 resident).
// Tiny vs. the main GEMM; removes all per-iteration VALU from the hot loop.
// ---------------------------------------------------------------------------
__global__ __launch_bounds__(256) void b2_norms_kernel(
    const float* __restrict__ desc2, float* __restrict__ b2, int B2)
{
    int row = blockIdx.x * blockDim.x + threadIdx.x;
    if (row >= B2) return;
    const v4f* p = (const v4f*)(desc2 + (size_t)row * DIM);
    float s = 0.f;
    #pragma unroll
    for (int k = 0; k < DIM / 4; ++k) {
        v4f v = p[k];
        s = fmaf(v.x, v.x, s);
        s = fmaf(v.y, v.y, s);
        s = fmaf(v.z, v.z, s);
        s = fmaf(v.w, v.w, s);
    }
    b2[row] = s;
}

// ---------------------------------------------------------------------------
// Pass 2: implicit GEMM with V_WMMA_F32_16X16X4_F32.
// Each wave owns 16 rows of desc1 (A tile register-resident in WMMA layout)
// and streams all of desc2 in 2x16-column tiles (two independent accumulator
// chains). Running key = ||b||^2 - 2*a.b per output element; ||a||^2 is added
// once at the end. HAS_B2=false falls back to fusing ||b||^2 into the loop.
// ---------------------------------------------------------------------------
template <bool HAS_B2>
__global__ __launch_bounds__(128) void nn_match_kernel(
    const float* __restrict__ desc1,
    const float* __restrict__ desc2,
    const float* __restrict__ b2arr,
    float* __restrict__ out,
    int B1, int B2)
{
    const int lane = threadIdx.x & 31;
    const int wave = threadIdx.x >> 5;
    const int rowBase = (blockIdx.x * 4 + wave) * 16;   // 16 rows per wave
    const int ln   = lane & 15;                         // 0..15
    const int hi   = lane >> 4;                         // 0 or 1 (K-half)
    const int koff = hi * 2;                            // K offset for this half

    // ---- Preload A tile (16 x 256) into registers in WMMA A-operand layout.
    v2f aReg[64];
    float asq = 0.f;
    {
        const float* ap = desc1 + (size_t)(rowBase + ln) * DIM + koff;
        #pragma unroll
        for (int kk = 0; kk < 64; ++kk) {
            v2f a = *(const v2f*)(ap + kk * 4);
            aReg[kk] = a;
            asq = fmaf(a.x, a.x, asq);
            asq = fmaf(a.y, a.y, asq);
        }
    }
    float a2full = asq + __shfl_xor(asq, 16, 32);       // ||desc1[rowBase+ln]||^2
    float a2v[8];
    #pragma unroll
    for (int r = 0; r < 8; ++r)
        a2v[r] = __shfl(a2full, hi * 8 + r, 32);        // uniform within a half

    float minv[8];   // running key = ||b||^2 - 2*a.b
    int   mini[8];
    #pragma unroll
    for (int r = 0; r < 8; ++r) { minv[r] = 3.4e38f; mini[r] = 0; }

    for (int col0 = 0; col0 < B2; col0 += 32) {
        const float* bp0 = desc2 + (size_t)(col0 + ln) * DIM + koff;
        const float* bp1 = bp0 + 16 * DIM;
        float b2f0, b2f1;
        if (HAS_B2) {                                   // issue early, consumed late
            b2f0 = b2arr[col0 + ln];
            b2f1 = b2arr[col0 + 16 + ln];
        }
        v8f acc0 = {};
        v8f acc1 = {};
        float bsq0 = 0.f, bsq1 = 0.f;
        #pragma unroll
        for (int kk = 0; kk < 64; ++kk) {
            v2f b0 = *(const v2f*)(bp0 + kk * 4);
            v2f b1 = *(const v2f*)(bp1 + kk * 4);
            acc0 = __builtin_amdgcn_wmma_f32_16x16x4_f32(
                false, aReg[kk], false, b0, (short)0, acc0, false, false);
            acc1 = __builtin_amdgcn_wmma_f32_16x16x4_f32(
                false, aReg[kk], false, b1, (short)0, acc1, false, false);
            if (!HAS_B2) {
                bsq0 = fmaf(b0.x, b0.x, bsq0); bsq0 = fmaf(b0.y, b0.y, bsq0);
                bsq1 = fmaf(b1.x, b1.x, bsq1); bsq1 = fmaf(b1.y, b1.y, bsq1);
            }
        }
        if (!HAS_B2) {
            b2f0 = bsq0 + __shfl_xor(bsq0, 16, 32);
            b2f1 = bsq1 + __shfl_xor(bsq1, 16, 32);
        }
        const int n0 = col0 + ln;
        const int n1 = col0 + 16 + ln;
        #pragma unroll
        for (int r = 0; r < 8; ++r) {
            float k0 = fmaf(-2.f, acc0[r], b2f0);
            float k1 = fmaf(-2.f, acc1[r], b2f1);
            if (k0 < minv[r]) { minv[r] = k0; mini[r] = n0; }
            if (k1 < minv[r]) { minv[r] = k1; mini[r] = n1; }
        }
    }

    // ---- Min/argmin across the 16 lanes of each half (butterfly stays in-half).
    #pragma unroll
    for (int r = 0; r < 8; ++r) {
        float v = minv[r];
        int   i = mini[r];
        #pragma unroll
        for (int off = 1; off < 16; off <<= 1) {
            float ov = __shfl_xor(v, off, 32);
            int   oi = __shfl_xor(i, off, 32);
            if (ov < v || (ov == v && oi < i)) { v = ov; i = oi; }
        }
        minv[r] = v; mini[r] = i;
    }

    // Lane 0 of each half writes rows 8*hi .. 8*hi+7 of this wave's block.
    if (ln == 0) {
        #pragma unroll
        for (int r = 0; r < 8; ++r) {
            int m = rowBase + hi * 8 + r;
            float d2 = a2v[r] + minv[r];                 // add ||a||^2 back
            out[m] = sqrtf(fmaxf(d2, 0.f));              // match_dists
            out[B1 + 2 * m + 0] = (float)m;              // idxs_in_1
            out[B1 + 2 * m + 1] = (float)mini[r];        // idxs_in_2
        }
    }
}

extern "C" void kernel_launch(void* const* d_in, const int* in_sizes, int n_in,
                              void* d_out, int out_size, void* d_ws, size_t ws_size,
                              hipStream_t stream) {
    const float* desc1 = (const float*)d_in[0];
    const float* desc2 = (const float*)d_in[1];
    float* out = (float*)d_out;

    const int B1 = in_sizes[0] / DIM;   // 16384
    const int B2 = in_sizes[1] / DIM;   // 16384
    const int blocks = B1 / 64;         // 4 waves/block * 16 rows/wave

    if (d_ws != nullptr && ws_size >= (size_t)B2 * sizeof(float)) {
        float* b2 = (float*)d_ws;
        b2_norms_kernel<<<(B2 + 255) / 256, 256, 0, stream>>>(desc2, b2, B2);
        nn_match_kernel<true><<<blocks, 128, 0, stream>>>(desc1, desc2, b2, out, B1, B2);
    } else {
        nn_match_kernel<false><<<blocks, 128, 0, stream>>>(desc1, desc2, nullptr, out, B1, B2);
    }
}